// Encoder_28140625723622
// MI455X (gfx1250) — compile-verified
//
#include <hip/hip_runtime.h>

typedef float v2f __attribute__((ext_vector_type(2)));
typedef float v8f __attribute__((ext_vector_type(8)));

#define D_FEAT 512
#define MT 5          // M-tiles per wave: 3125 tiles = 625 * 5 (exact)

// ---------------- init kernels ----------------

__global__ void zero_f32_kernel(float* __restrict__ p, long long n4) {
  long long i = (long long)blockIdx.x * blockDim.x + threadIdx.x;
  if (i < n4) ((float4*)p)[i] = make_float4(0.f, 0.f, 0.f, 0.f);
}

__global__ void zero_flags_kernel(int* __restrict__ flags, int n) {
  int i = blockIdx.x * blockDim.x + threadIdx.x;
  if (i < n) flags[i] = 0;
}

__global__ void set_mask_kernel(const int* __restrict__ mask_idx, int* __restrict__ flags,
                                float* __restrict__ out_tail, int n_mask) {
  int i = blockIdx.x * blockDim.x + threadIdx.x;
  if (i < n_mask) {
    int m = mask_idx[i];
    flags[m] = 1;
    out_tail[i] = (float)m;   // second tuple output, flattened after `out`
  }
}

// ---------------- fp32 WMMA GEMM: x = feat @ W (unmasked) ----------------
// Masking is applied downstream: a masked row of A only zeroes the matching
// row of x, and x is consumed solely via x[edge_src] -> drop those edges.
//
// One wave computes an 80x16 strip (5 stacked 16x16 tiles, one B fragment
// shared per k-step). 128 k-steps of V_WMMA_F32_16X16X4_F32 -> 640 WMMA/wave.
// Exact tiling (50000 = 625*80, 512 = 32*16): uniform control flow, EXEC
// all-1s at every WMMA.
//
// A-frag (16x4 f32): lane L holds A[m0 + L%16][k + (L/16)*2 + {0,1}]  (b64 load)
// B-frag (4x16 f32): lane L holds B[k + (L/16)*2 + {0,1}][n0 + L%16]
// C/D (16x16 f32, 8 VGPRs): VGPR j -> M = m0 + (L/16)*8 + j, N = n0 + L%16
__global__ void __launch_bounds__(128)
gemm_wmma_f32_kernel(const float* __restrict__ feat, const float* __restrict__ W,
                     float* __restrict__ x) {
  const int lane  = threadIdx.x & 31;
  const int wave  = threadIdx.x >> 5;     // 4 waves / block, one N-tile each
  const int mg    = blockIdx.x >> 3;      // 0..624 : 80-row M-group
  const int ng    = blockIdx.x & 7;       // group of 4 N-tiles
  const int nt    = ng * 4 + wave;        // 0..31
  const int m0    = mg * (16 * MT);
  const int n0    = nt * 16;

  const int khalf = (lane >> 4) << 1;     // 0 or 2
  const int ncol  = n0 + (lane & 15);

  // A base: row (m0 + lane%16), column khalf; tile t adds t*16 rows = t*32KB
  const float* abase = feat + (long long)(m0 + (lane & 15)) * D_FEAT + khalf;
  // B base: row khalf, column ncol; advances 4 rows (8KB) per k-step
  const float* wp    = W + (long long)khalf * D_FEAT + ncol;

  v8f acc[MT];
#pragma unroll
  for (int t = 0; t < MT; ++t) acc[t] = (v8f){0.f,0.f,0.f,0.f,0.f,0.f,0.f,0.f};

#pragma unroll 4
  for (int k = 0; k < D_FEAT; k += 4) {
    v2f b;
    b.x = wp[0];
    b.y = wp[D_FEAT];
    wp += 4 * D_FEAT;
#pragma unroll
    for (int t = 0; t < MT; ++t) {
      v2f a = *(const v2f*)(abase + (long long)t * 16 * D_FEAT + k);  // b64, imm offset
      acc[t] = __builtin_amdgcn_wmma_f32_16x16x4_f32(
          false, a, false, b, (short)0, acc[t], false, false);
    }
  }

  // store: 5 tiles x 8 rows, all immediate offsets off one address
  float* xout = x + (long long)(m0 + ((lane >> 4) << 3)) * D_FEAT + ncol;
#pragma unroll
  for (int t = 0; t < MT; ++t)
#pragma unroll
    for (int j = 0; j < 8; ++j)
      xout[(long long)(t * 16 + j) * D_FEAT] = acc[t][j];
}

// ---------------- edge gather + scatter-add ----------------
// One block per edge, 128 threads x float4 (b128 gather, 4 f32 atomics each).
// Masked sources contribute exactly zero in the reference -> uniform early
// exit skips ~MASK_RATE of all gather + atomic traffic.
// x (102 MB) is L2-resident (192 MB L2), so gathers hit L2; atomics are L2 RMW.
__global__ void __launch_bounds__(128)
edge_scatter_kernel(const float* __restrict__ x, const float* __restrict__ edge_val,
                    const int* __restrict__ edge_src, const int* __restrict__ edge_dst,
                    const int* __restrict__ flags, float* __restrict__ out, int n_edges) {
  int e = blockIdx.x;
  if (e >= n_edges) return;
  int s = edge_src[e];
  if (flags[s]) return;                 // masked source: message is all-zero
  int d   = edge_dst[e];
  float v = edge_val[e];
  int c   = threadIdx.x * 4;
  float4 xv = *(const float4*)(x + (long long)s * D_FEAT + c);
  float* o = out + (long long)d * D_FEAT + c;
  atomicAdd(o,     v * xv.x);
  atomicAdd(o + 1, v * xv.y);
  atomicAdd(o + 2, v * xv.z);
  atomicAdd(o + 3, v * xv.w);
}

// ---------------- launcher ----------------

extern "C" void kernel_launch(void* const* d_in, const int* in_sizes, int n_in,
                              void* d_out, int out_size, void* d_ws, size_t ws_size,
                              hipStream_t stream) {
  const float* feat     = (const float*)d_in[0];
  const float* weight   = (const float*)d_in[1];
  const float* edge_val = (const float*)d_in[2];
  const int*   edge_src = (const int*)d_in[3];
  const int*   edge_dst = (const int*)d_in[4];
  const int*   mask_idx = (const int*)d_in[5];

  const int n_nodes = in_sizes[0] / D_FEAT;  // 50000
  const int n_edges = in_sizes[2];           // 400000
  const int n_mask  = in_sizes[5];           // 15000

  float* out      = (float*)d_out;
  float* out_tail = out + (long long)n_nodes * D_FEAT;

  // workspace: x [n_nodes*512 f32] then flags [n_nodes i32]
  float* x   = (float*)d_ws;
  int* flags = (int*)((char*)d_ws + (size_t)n_nodes * D_FEAT * sizeof(float));

  const long long n_out = (long long)n_nodes * D_FEAT;   // 25,600,000 (divisible by 4)
  const long long n4    = n_out / 4;

  zero_f32_kernel<<<(unsigned)((n4 + 255) / 256), 256, 0, stream>>>(out, n4);
  zero_flags_kernel<<<(n_nodes + 255) / 256, 256, 0, stream>>>(flags, n_nodes);
  set_mask_kernel<<<(n_mask + 255) / 256, 256, 0, stream>>>(mask_idx, flags, out_tail, n_mask);

  const int mgroups = n_nodes / (16 * MT);   // 625 (exact)
  gemm_wmma_f32_kernel<<<mgroups * 8, 128, 0, stream>>>(feat, weight, x);

  edge_scatter_kernel<<<n_edges, 128, 0, stream>>>(x, edge_val, edge_src, edge_dst,
                                                   flags, out, n_edges);
}